// GCN_44461501448279
// MI455X (gfx1250) — compile-verified
//
#include <hip/hip_runtime.h>

#define FIN 256
#define HID 64
#define COUT 40

typedef float v2f __attribute__((ext_vector_type(2)));
typedef float v8f __attribute__((ext_vector_type(8)));

// ---------------- degree / dinv ----------------
__global__ __launch_bounds__(256) void k_deg_init(float* __restrict__ deg, int n) {
    int i = blockIdx.x * 256 + threadIdx.x;
    if (i < n) deg[i] = 1.0f;  // self loop contributes 1
}

__global__ __launch_bounds__(256) void k_deg_acc(const int* __restrict__ dst, int E,
                                                 float* __restrict__ deg) {
    int i = blockIdx.x * 256 + threadIdx.x;
    if (i < E) atomicAdd(deg + dst[i], 1.0f);
}

__global__ __launch_bounds__(256) void k_dinv(float* __restrict__ deg, int n) {
    int i = blockIdx.x * 256 + threadIdx.x;
    if (i < n) deg[i] = rsqrtf(deg[i]);
}

// zero n float4s
__global__ __launch_bounds__(256) void k_zero4(float4* __restrict__ p, int n4) {
    int i = blockIdx.x * 256 + threadIdx.x;
    if (i < n4) p[i] = make_float4(0.f, 0.f, 0.f, 0.f);
}

// ---------------- GEMM1: H[N,64] = X[N,256] @ W1[256,64] (fp32 WMMA) ----------------
__global__ __launch_bounds__(256) void k_gemm1(const float* __restrict__ X,
                                               const float* __restrict__ W,
                                               float* __restrict__ H, int nrows) {
    int wave = (blockIdx.x * 256 + threadIdx.x) >> 5;
    int lane = threadIdx.x & 31;
    int m0 = wave << 4;              // 16-row strip per wave (nrows % 16 == 0)
    if (m0 >= nrows) return;         // wave-uniform: EXEC stays all-ones for WMMA
    int row = lane & 15;
    int kk  = (lane >> 4) << 1;      // K offset 0 or 2 per half-wave
    const float* Xr = X + (size_t)(m0 + row) * FIN + kk;
    v8f c0 = {}, c1 = {}, c2 = {}, c3 = {};
    for (int k0 = 0; k0 < FIN; k0 += 4) {
        v2f a;
        a.x = Xr[k0];
        a.y = Xr[k0 + 1];
        const float* Wp = W + (size_t)(k0 + kk) * HID;
        v2f b0 = { Wp[row],      Wp[HID + row] };
        v2f b1 = { Wp[16 + row], Wp[HID + 16 + row] };
        v2f b2 = { Wp[32 + row], Wp[HID + 32 + row] };
        v2f b3 = { Wp[48 + row], Wp[HID + 48 + row] };
        c0 = __builtin_amdgcn_wmma_f32_16x16x4_f32(false, a, false, b0, (short)0, c0, false, false);
        c1 = __builtin_amdgcn_wmma_f32_16x16x4_f32(false, a, false, b1, (short)0, c1, false, false);
        c2 = __builtin_amdgcn_wmma_f32_16x16x4_f32(false, a, false, b2, (short)0, c2, false, false);
        c3 = __builtin_amdgcn_wmma_f32_16x16x4_f32(false, a, false, b3, (short)0, c3, false, false);
    }
    int mrow = m0 + ((lane >> 4) << 3);  // lanes 16-31 hold M=8..15
    int col  = lane & 15;
    for (int r = 0; r < 8; ++r) {
        float* out = H + (size_t)(mrow + r) * HID + col;
        out[0]  = c0[r];
        out[16] = c1[r];
        out[32] = c2[r];
        out[48] = c3[r];
    }
}

// ---------------- GEMM2: H2[N,40] = H1[N,64] @ W2[64,40] (fp32 WMMA, col guard) ----------------
__global__ __launch_bounds__(256) void k_gemm2(const float* __restrict__ H1,
                                               const float* __restrict__ W,
                                               float* __restrict__ H2, int nrows) {
    int wave = (blockIdx.x * 256 + threadIdx.x) >> 5;
    int lane = threadIdx.x & 31;
    int m0 = wave << 4;
    if (m0 >= nrows) return;
    int row = lane & 15;
    int kk  = (lane >> 4) << 1;
    const float* Hr = H1 + (size_t)(m0 + row) * HID + kk;
    v8f c0 = {}, c1 = {}, c2 = {};
    for (int k0 = 0; k0 < HID; k0 += 4) {
        v2f a;
        a.x = Hr[k0];
        a.y = Hr[k0 + 1];
        const float* Wp = W + (size_t)(k0 + kk) * COUT;
        v2f b0 = { Wp[row],      Wp[COUT + row] };
        v2f b1 = { Wp[16 + row], Wp[COUT + 16 + row] };
        int col2 = 32 + row;
        v2f b2 = { (col2 < COUT) ? Wp[col2] : 0.0f,
                   (col2 < COUT) ? Wp[COUT + col2] : 0.0f };
        c0 = __builtin_amdgcn_wmma_f32_16x16x4_f32(false, a, false, b0, (short)0, c0, false, false);
        c1 = __builtin_amdgcn_wmma_f32_16x16x4_f32(false, a, false, b1, (short)0, c1, false, false);
        c2 = __builtin_amdgcn_wmma_f32_16x16x4_f32(false, a, false, b2, (short)0, c2, false, false);
    }
    int mrow = m0 + ((lane >> 4) << 3);
    int col  = lane & 15;
    for (int r = 0; r < 8; ++r) {
        float* out = H2 + (size_t)(mrow + r) * COUT;
        out[col]      = c0[r];
        out[16 + col] = c1[r];
        if (32 + col < COUT) out[32 + col] = c2[r];
    }
}

// ---------------- layer-1 edge scatter: 16 threads/edge, float4 gather ----------------
// agg[dst, f..f+3] += h[src, f..f+3] * dinv[src]*dinv[dst]
__global__ __launch_bounds__(256) void k_agg1(const int* __restrict__ src,
                                              const int* __restrict__ dst, int E,
                                              const float* __restrict__ dinv,
                                              const float4* __restrict__ H4,
                                              float* __restrict__ A) {
    long long idx = (long long)blockIdx.x * 256 + threadIdx.x;
    if (idx >= (long long)E * 16) return;
    int e  = (int)(idx >> 4);
    int f4 = (int)(idx & 15);            // which float4 of the 64-feature row
    int s = src[e], d = dst[e];
    float nrm = dinv[s] * dinv[d];
    float4 v = H4[(size_t)s * (HID / 4) + f4];
    float* out = A + (size_t)d * HID + f4 * 4;
    atomicAdd(out + 0, v.x * nrm);
    atomicAdd(out + 1, v.y * nrm);
    atomicAdd(out + 2, v.z * nrm);
    atomicAdd(out + 3, v.w * nrm);
}

// finalize layer 1 (float4): A = relu(A + H*dinv^2 + b1), in place
__global__ __launch_bounds__(256) void k_fin1(float4* __restrict__ A,
                                              const float4* __restrict__ H4,
                                              const float* __restrict__ dinv,
                                              const float* __restrict__ b1, int n) {
    int idx = blockIdx.x * 256 + threadIdx.x;  // n * 16 float4s
    if (idx >= n * (HID / 4)) return;
    int i  = idx >> 4;
    int f4 = idx & 15;
    float di = dinv[i];
    float nn = di * di;
    float4 a = A[idx];
    float4 h = H4[idx];
    const float* b = b1 + f4 * 4;
    a.x = fmaxf(a.x + h.x * nn + b[0], 0.f);
    a.y = fmaxf(a.y + h.y * nn + b[1], 0.f);
    a.z = fmaxf(a.z + h.z * nn + b[2], 0.f);
    a.w = fmaxf(a.w + h.w * nn + b[3], 0.f);
    A[idx] = a;
}

// ---------------- layer-2 edge scatter: 10 threads/edge, float4 (40 = 10*4) ----------------
__global__ __launch_bounds__(320) void k_agg2(const int* __restrict__ src,
                                              const int* __restrict__ dst, int E,
                                              const float* __restrict__ dinv,
                                              const float4* __restrict__ H4,
                                              float* __restrict__ out) {
    int t = threadIdx.x;                  // 320 threads = 32 edges * 10
    int e  = blockIdx.x * 32 + t / 10;
    int f4 = t % 10;
    if (e >= E) return;
    int s = src[e], d = dst[e];
    float nrm = dinv[s] * dinv[d];
    float4 v = H4[(size_t)s * (COUT / 4) + f4];
    float* o = out + (size_t)d * COUT + f4 * 4;
    atomicAdd(o + 0, v.x * nrm);
    atomicAdd(o + 1, v.y * nrm);
    atomicAdd(o + 2, v.z * nrm);
    atomicAdd(o + 3, v.w * nrm);
}

// finalize layer 2 (float4): out += H2*dinv^2 + b2
__global__ __launch_bounds__(256) void k_fin2(float4* __restrict__ out,
                                              const float4* __restrict__ H4,
                                              const float* __restrict__ dinv,
                                              const float* __restrict__ b2, int n) {
    int idx = blockIdx.x * 256 + threadIdx.x;  // n * 10 float4s
    if (idx >= n * (COUT / 4)) return;
    int i  = idx / 10;
    int f4 = idx % 10;
    float di = dinv[i];
    float nn = di * di;
    float4 o = out[idx];
    float4 h = H4[idx];
    const float* b = b2 + f4 * 4;
    o.x += h.x * nn + b[0];
    o.y += h.y * nn + b[1];
    o.z += h.z * nn + b[2];
    o.w += h.w * nn + b[3];
    out[idx] = o;
}

extern "C" void kernel_launch(void* const* d_in, const int* in_sizes, int n_in,
                              void* d_out, int out_size, void* d_ws, size_t ws_size,
                              hipStream_t stream) {
    const float* x  = (const float*)d_in[0];
    const float* W1 = (const float*)d_in[1];
    const float* b1 = (const float*)d_in[2];
    const float* W2 = (const float*)d_in[3];
    const float* b2 = (const float*)d_in[4];
    const int*   ei = (const int*)d_in[5];

    const int Nn = in_sizes[0] / FIN;       // 100000 (multiple of 16)
    const int E  = in_sizes[5] / 2;         // 1,600,000
    const int* src = ei;
    const int* dst = ei + E;

    float* ws   = (float*)d_ws;
    float* dinv = ws;                       // N floats (deg, then rsqrt in place)
    float* h    = ws + Nn;                  // N*64 (reused as h2 N*40 once dead)
    float* agg  = h + (size_t)Nn * HID;     // N*64
    float* h2   = h;                        // alias: h dead after k_fin1
    float* out  = (float*)d_out;

    int gN  = (Nn + 255) / 256;
    int gE  = (E + 255) / 256;
    int h4n = Nn * (HID / 4);               // N*16 float4s
    int o4n = Nn * (COUT / 4);              // N*10 float4s
    int gH4 = (h4n + 255) / 256;
    int gO4 = (o4n + 255) / 256;
    int gE1 = (int)(((long long)E * 16 + 255) / 256);
    int gE2 = (E + 31) / 32;
    int gemmBlocks = ((Nn / 16) + 7) / 8;   // 8 waves per 256-thread block

    // degree -> dinv
    k_deg_init<<<gN, 256, 0, stream>>>(dinv, Nn);
    k_deg_acc<<<gE, 256, 0, stream>>>(dst, E, dinv);
    k_dinv<<<gN, 256, 0, stream>>>(dinv, Nn);

    // layer 1
    k_gemm1<<<gemmBlocks, 256, 0, stream>>>(x, W1, h, Nn);
    k_zero4<<<gH4, 256, 0, stream>>>((float4*)agg, h4n);
    k_agg1<<<gE1, 256, 0, stream>>>(src, dst, E, dinv, (const float4*)h, agg);
    k_fin1<<<gH4, 256, 0, stream>>>((float4*)agg, (const float4*)h, dinv, b1, Nn);

    // layer 2
    k_gemm2<<<gemmBlocks, 256, 0, stream>>>(agg, W2, h2, Nn);
    k_zero4<<<gO4, 256, 0, stream>>>((float4*)out, o4n);
    k_agg2<<<gE2, 320, 0, stream>>>(src, dst, E, dinv, (const float4*)h2, out);
    k_fin2<<<gO4, 256, 0, stream>>>((float4*)out, (const float4*)h2, dinv, b2, Nn);
}